// Demo_42271068127574
// MI455X (gfx1250) — compile-verified
//
#include <hip/hip_runtime.h>

#define DD 128
#define D4 32           // DD/4 float4 per row
#define RCOF 0.5f

typedef float v2f __attribute__((ext_vector_type(2)));
typedef float v8f __attribute__((ext_vector_type(8)));

// ---------------------------------------------------------------------------
// degree count (symmetric bipartite graph): cnt[u]++ , cnt[v+na]++
// ---------------------------------------------------------------------------
__global__ void k_degree(const int* __restrict__ eu, const int* __restrict__ ev,
                         int E, int na, unsigned* __restrict__ cnt) {
  int stride = gridDim.x * blockDim.x;
  for (int e = blockIdx.x * blockDim.x + threadIdx.x; e < E; e += stride) {
    atomicAdd(&cnt[eu[e]], 1u);
    atomicAdd(&cnt[ev[e] + na], 1u);
  }
}

// cnt (uint) -> dinv (float) in place:  1 / (sqrt(deg) + 1e-8)
__global__ void k_dinv(float* __restrict__ dinv, int n) {
  int i = blockIdx.x * blockDim.x + threadIdx.x;
  if (i < n) {
    unsigned c = ((const unsigned*)dinv)[i];
    dinv[i] = 1.0f / (sqrtf((float)c) + 1e-8f);
  }
}

// ---------------------------------------------------------------------------
// feats = concat(a,b); acc = feats   (float4 streaming)
// ---------------------------------------------------------------------------
__global__ void k_concat(const float4* __restrict__ a, const float4* __restrict__ b,
                         long long na4, long long n4,
                         float4* __restrict__ feats, float4* __restrict__ acc) {
  long long stride = (long long)gridDim.x * blockDim.x;
  for (long long i = (long long)blockIdx.x * blockDim.x + threadIdx.x; i < n4; i += stride) {
    float4 v = (i < na4) ? a[i] : b[i - na4];
    feats[i] = v;
    acc[i]   = v;
  }
}

// ---------------------------------------------------------------------------
// Row L2 norms via V_WMMA_F32_16X16X4_F32: one wave handles a 16-row tile,
// accumulates Gram = X * X^T over K=128 in 32 k-steps, reads the diagonal.
// A 16x4 f32 fragment layout (ISA 7.12.2): lanes 0-15 hold M=lane, K={0,1};
// lanes 16-31 hold M=lane-16, K={2,3}. B uses the mirrored layout, so the
// same per-lane fragment serves both operands (B = A^T role).
// rnorm[r] = 1 / max(||x_r||, 1e-12)
// ---------------------------------------------------------------------------
__global__ void k_rownorms_wmma(const float* __restrict__ X, int nrows,
                                float* __restrict__ rnorm) {
  const int lane = threadIdx.x & 31;
  int tile = blockIdx.x * (blockDim.x >> 5) + (threadIdx.x >> 5);
  int ntiles = (nrows + 15) >> 4;
  if (tile >= ntiles) return;                 // wave-uniform: EXEC stays all-1s
  const int row0 = tile << 4;
  const int mA = lane & 15;
  const int kA = (lane >> 4) << 1;            // 0 or 2
  int rA = row0 + mA;
  if (rA >= nrows) rA = nrows - 1;            // clamp tail (loads stay uniform)
  const float* __restrict__ Arow = X + (size_t)rA * DD;

  v8f c = {};
#pragma unroll
  for (int kk = 0; kk < DD; kk += 4) {
    v2f a;
    a.x = Arow[kk + kA];
    a.y = Arow[kk + kA + 1];
    // D = A * A^T(mirrored) + C    -> Gram of the 16-row tile
    c = __builtin_amdgcn_wmma_f32_16x16x4_f32(false, a, false, a, (short)0, c,
                                              false, false);
  }

  // C/D layout: VGPR j -> lanes0-15:(M=j,N=lane), lanes16-31:(M=j+8,N=lane-16).
  // Diagonal (M==N): lane j (j<8) holds c[j] for row j; lane j+24 holds c[j]
  // for row j+8.
  bool isDiag = (lane < 8) || (lane >= 24);
  int j      = (lane < 8) ? lane : (lane - 24);
  int outRow = (lane < 8) ? lane : (lane - 16);
  float d = c[0];
#pragma unroll
  for (int t = 1; t < 8; ++t) d = (j == t) ? c[t] : d;
  int r = row0 + outRow;
  if (isDiag && r < nrows) {
    rnorm[r] = 1.0f / fmaxf(sqrtf(d), 1e-12f);
  }
}

// ini[i] = feats[i] * rnorm[row]
__global__ void k_scale_ini(const float4* __restrict__ feats,
                            const float* __restrict__ rnorm, long long n4,
                            float4* __restrict__ ini) {
  long long stride = (long long)gridDim.x * blockDim.x;
  for (long long i = (long long)blockIdx.x * blockDim.x + threadIdx.x; i < n4; i += stride) {
    float s = rnorm[i >> 5];
    float4 v = feats[i];
    ini[i] = make_float4(v.x * s, v.y * s, v.z * s, v.w * s);
  }
}

// tmp += RC * ini
__global__ void k_axpy(float4* __restrict__ tmp, const float4* __restrict__ ini,
                       long long n4) {
  long long stride = (long long)gridDim.x * blockDim.x;
  for (long long i = (long long)blockIdx.x * blockDim.x + threadIdx.x; i < n4; i += stride) {
    float4 t = tmp[i], u = ini[i];
    tmp[i] = make_float4(t.x + RCOF * u.x, t.y + RCOF * u.y,
                         t.z + RCOF * u.z, t.w + RCOF * u.w);
  }
}

// v = tmp * rnorm[row]; feats = v; acc += v
__global__ void k_scale_acc(const float4* __restrict__ tmp,
                            const float* __restrict__ rnorm, long long n4,
                            float4* __restrict__ feats, float4* __restrict__ acc) {
  long long stride = (long long)gridDim.x * blockDim.x;
  for (long long i = (long long)blockIdx.x * blockDim.x + threadIdx.x; i < n4; i += stride) {
    float s = rnorm[i >> 5];
    float4 t = tmp[i];
    float4 v = make_float4(t.x * s, t.y * s, t.z * s, t.w * s);
    feats[i] = v;
    float4 a = acc[i];
    acc[i] = make_float4(a.x + v.x, a.y + v.y, a.z + v.z, a.w + v.w);
  }
}

__global__ void k_copy(const float4* __restrict__ src, float4* __restrict__ dst,
                       long long n4) {
  long long stride = (long long)gridDim.x * blockDim.x;
  for (long long i = (long long)blockIdx.x * blockDim.x + threadIdx.x; i < n4; i += stride)
    dst[i] = src[i];
}

// ---------------------------------------------------------------------------
// Symmetric-normalized COO SPMM: one wave per directed edge.
// 32 lanes x float4 = one full 512B feature row, coalesced gather + fp32
// hardware atomic scatter (rows live in 192MB L2). Prefetch next gather row.
// ---------------------------------------------------------------------------
__global__ void k_spmm_sym(const int* __restrict__ eu, const int* __restrict__ ev,
                           long long E, int na, const float* __restrict__ dinv,
                           const float* __restrict__ feats, float* __restrict__ outp) {
  const int lane = threadIdx.x & 31;
  long long w = (long long)blockIdx.x * (blockDim.x >> 5) + (threadIdx.x >> 5);
  long long stride = (long long)gridDim.x * (blockDim.x >> 5);
  const long long E2 = 2 * E;
  for (long long e = w; e < E2; e += stride) {
    int r, c;
    if (e < E) { r = eu[e];     c = ev[e] + na; }
    else       { r = ev[e - E] + na; c = eu[e - E]; }
    long long en = e + stride;                 // prefetch next gather row
    if (en < E2) {
      int cn = (en < E) ? (ev[en] + na) : eu[en - E];
      __builtin_prefetch(feats + (size_t)cn * DD + (size_t)lane * 4, 0, 1);
    }
    float val = dinv[r] * dinv[c];
    float4 x = ((const float4*)(feats + (size_t)c * DD))[lane];
    float* dst = outp + (size_t)r * DD + (size_t)lane * 4;
    unsafeAtomicAdd(dst + 0, val * x.x);
    unsafeAtomicAdd(dst + 1, val * x.y);
    unsafeAtomicAdd(dst + 2, val * x.z);
    unsafeAtomicAdd(dst + 3, val * x.w);
  }
}

// Weighted COO SPMM (BI aggregation), accumulates directly into d_out region.
__global__ void k_spmm_bi(const int* __restrict__ eb, const int* __restrict__ ei,
                          const float* __restrict__ vals, long long E,
                          const float* __restrict__ src, float* __restrict__ dstBase) {
  const int lane = threadIdx.x & 31;
  long long w = (long long)blockIdx.x * (blockDim.x >> 5) + (threadIdx.x >> 5);
  long long stride = (long long)gridDim.x * (blockDim.x >> 5);
  for (long long e = w; e < E; e += stride) {
    int r = eb[e], c = ei[e];
    long long en = e + stride;
    if (en < E) __builtin_prefetch(src + (size_t)ei[en] * DD + (size_t)lane * 4, 0, 1);
    float v = vals[e];
    float4 x = ((const float4*)(src + (size_t)c * DD))[lane];
    float* dst = dstBase + (size_t)r * DD + (size_t)lane * 4;
    unsafeAtomicAdd(dst + 0, v * x.x);
    unsafeAtomicAdd(dst + 1, v * x.y);
    unsafeAtomicAdd(dst + 2, v * x.z);
    unsafeAtomicAdd(dst + 3, v * x.w);
  }
}

// ---------------------------------------------------------------------------
extern "C" void kernel_launch(void* const* d_in, const int* in_sizes, int n_in,
                              void* d_out, int out_size, void* d_ws, size_t ws_size,
                              hipStream_t stream) {
  (void)n_in; (void)out_size; (void)ws_size;
  const float* users   = (const float*)d_in[0];
  const float* bundles = (const float*)d_in[1];
  const float* items   = (const float*)d_in[2];
  const float* bi_vals = (const float*)d_in[3];
  const int* ub_u = (const int*)d_in[4];
  const int* ub_b = (const int*)d_in[5];
  const int* ui_u = (const int*)d_in[6];
  const int* ui_i = (const int*)d_in[7];
  const int* bi_b = (const int*)d_in[8];
  const int* bi_i = (const int*)d_in[9];

  const int NU = in_sizes[0] / DD;
  const int NB = in_sizes[1] / DD;
  const int NI = in_sizes[2] / DD;
  const long long E_BI = in_sizes[3];
  const long long E_UB = in_sizes[4];
  const long long E_UI = in_sizes[6];

  const int N1 = NU + NB, N2 = NU + NI;
  const int NM = (N1 > N2) ? N1 : N2;

  float* feats = (float*)d_ws;
  float* ini   = feats + (size_t)NM * DD;
  float* acc   = ini   + (size_t)NM * DD;
  float* tmp   = acc   + (size_t)NM * DD;
  float* rnorm = tmp   + (size_t)NM * DD;
  float* dinv  = rnorm + NM;
  float* out   = (float*)d_out;

  const int TPB = 256;
  auto blocks = [](long long work, int per) {
    long long g = (work + per - 1) / per;
    if (g < 1) g = 1;
    if (g > 1048576) g = 1048576;
    return (int)g;
  };

  auto propagate = [&](const float* afeat, const float* bfeat, int na, int nb,
                       const int* eu, const int* ev, long long E) {
    const int n = na + nb;
    const long long n4 = (long long)n * D4;
    const int ntiles = (n + 15) / 16;
    // degrees -> dinv
    hipMemsetAsync(dinv, 0, sizeof(unsigned) * (size_t)n, stream);
    k_degree<<<blocks(E, TPB), TPB, 0, stream>>>(eu, ev, (int)E, na, (unsigned*)dinv);
    k_dinv<<<blocks(n, TPB), TPB, 0, stream>>>(dinv, n);
    // feats = concat, acc = feats
    k_concat<<<blocks(n4, TPB), TPB, 0, stream>>>((const float4*)afeat, (const float4*)bfeat,
                                                  (long long)na * D4, n4,
                                                  (float4*)feats, (float4*)acc);
    // ini = l2norm(feats)
    k_rownorms_wmma<<<(ntiles + 7) / 8, TPB, 0, stream>>>(feats, n, rnorm);
    k_scale_ini<<<blocks(n4, TPB), TPB, 0, stream>>>((const float4*)feats, rnorm, n4,
                                                     (float4*)ini);
    for (int layer = 0; layer < 2; ++layer) {
      hipMemsetAsync(tmp, 0, sizeof(float) * (size_t)n * DD, stream);
      // tmp = A_hat * feats       (one wave per directed edge)
      k_spmm_sym<<<blocks(2 * E, 8), TPB, 0, stream>>>(eu, ev, E, na, dinv, feats, tmp);
      // tmp += RC*ini ; the (layer+2) divisor cancels under L2 normalization
      k_axpy<<<blocks(n4, TPB), TPB, 0, stream>>>((float4*)tmp, (const float4*)ini, n4);
      k_rownorms_wmma<<<(ntiles + 7) / 8, TPB, 0, stream>>>(tmp, n, rnorm);
      // feats = l2norm(tmp) ; acc += feats
      k_scale_acc<<<blocks(n4, TPB), TPB, 0, stream>>>((const float4*)tmp, rnorm, n4,
                                                       (float4*)feats, (float4*)acc);
    }
  };

  const long long nu4 = (long long)NU * D4;
  const long long nb4 = (long long)NB * D4;

  // ---- UB (history) propagation ----
  propagate(users, bundles, NU, NB, ub_u, ub_b, E_UB);
  // hist_users -> rows [NU,2NU) ; hist_bundles -> rows [2NU+NB, 2NU+2NB)
  k_copy<<<blocks(nu4, TPB), TPB, 0, stream>>>((const float4*)acc,
                                               (float4*)out + nu4, nu4);
  k_copy<<<blocks(nb4, TPB), TPB, 0, stream>>>((const float4*)acc + nu4,
                                               (float4*)out + 2 * nu4 + nb4, nb4);

  // ---- UI (affinity) propagation ----
  propagate(users, items, NU, NI, ui_u, ui_i, E_UI);
  // aff_users -> rows [0,NU)
  k_copy<<<blocks(nu4, TPB), TPB, 0, stream>>>((const float4*)acc, (float4*)out, nu4);

  // ---- BI aggregation into aff_bundles rows [2NU, 2NU+NB) ----
  float* outB = out + (size_t)2 * NU * DD;
  hipMemsetAsync(outB, 0, sizeof(float) * (size_t)NB * DD, stream);
  k_spmm_bi<<<blocks(E_BI, 8), TPB, 0, stream>>>(bi_b, bi_i, bi_vals, E_BI,
                                                 acc + (size_t)NU * DD, outB);
}